// FRGAT_15229954031688
// MI455X (gfx1250) — compile-verified
//
#include <hip/hip_runtime.h>
#include <math.h>

typedef __attribute__((ext_vector_type(2))) float v2f;
typedef __attribute__((ext_vector_type(8))) float v8f;

#define D_DIM 100
#define D_VEC 25            // 100 floats = 25 float4
#define ALPHA 0.2f

// ---------------------------------------------------------------------------
// Zero-init output + softmax statistics (harness poisons buffers with 0xAA).
// ---------------------------------------------------------------------------
__global__ void frgat_init(float* __restrict__ out, int n_out,
                           float* __restrict__ denom,
                           unsigned* __restrict__ seg_max, int n_ent) {
    int i = blockIdx.x * blockDim.x + threadIdx.x;
    if (i < n_out) out[i] = 0.0f;
    if (i < n_ent) { denom[i] = 0.0f; seg_max[i] = 0u; }  // 0u == bits of +0.0f
}

// ---------------------------------------------------------------------------
// Y[nrows x 100] = X[nrows x 100] @ W[100 x 100] using V_WMMA_F32_16X16X4_F32.
// One wave per 16x16 output tile; K-loop in steps of 4 (100 % 4 == 0).
// Out-of-range rows/cols are CLAMPED on load (branch-free inner loop: pure
// global_load + v_wmma chain) and guarded only on store.
// Wave32 fragment layouts per CDNA5 ISA 7.12.2:
//   A (16x4 f32, 2 VGPR): lanes 0-15 row M=l hold K=0,1; lanes 16-31 hold K=2,3
//   B (4x16 f32, 2 VGPR): lanes 0-15 col N=l hold K=0,1; lanes 16-31 hold K=2,3
//   C/D (16x16 f32, 8 VGPR): vgpr r, lanes 0-15 -> M=r; lanes 16-31 -> M=r+8
// ---------------------------------------------------------------------------
__global__ void proj_wmma(const float* __restrict__ X,
                          const float* __restrict__ W,
                          float* __restrict__ Y, int nrows) {
    const int lane     = threadIdx.x & 31;
    const int wave     = (blockIdx.x * blockDim.x + threadIdx.x) >> 5;
    const int colTiles = (D_DIM + 15) / 16;               // 7
    const int rowTile  = wave / colTiles;
    const int colTile  = wave % colTiles;
    const int row0     = rowTile * 16;
    const int col0     = colTile * 16;
    if (row0 >= nrows) return;                            // uniform per wave

    const int half = lane >> 4;                           // 0 or 1
    const int l    = lane & 15;

    // Clamp: rows/cols beyond the edge read a valid neighbor; never stored.
    const int arow = (row0 + l < nrows) ? (row0 + l) : (nrows - 1);
    const int bcol = (col0 + l < D_DIM) ? (col0 + l) : (D_DIM - 1);

    const float* __restrict__ pa = X + (long)arow * D_DIM + half * 2;
    const float* __restrict__ pb = W + bcol + half * 2 * D_DIM;

    v8f acc = {};
#pragma unroll
    for (int k = 0; k < D_DIM; k += 4) {
        v2f a, b;
        a.x = pa[k];                 // consecutive pair -> global_load_b64
        a.y = pa[k + 1];
        b.x = pb[(long)k * D_DIM];   // coalesced across lanes (stride 1)
        b.y = pb[(long)(k + 1) * D_DIM];
        acc = __builtin_amdgcn_wmma_f32_16x16x4_f32(
            /*neg_a=*/false, a, /*neg_b=*/false, b,
            /*c_mod=*/(short)0, acc, /*reuse_a=*/false, /*reuse_b=*/false);
    }

    const int ccol = col0 + l;
#pragma unroll
    for (int r = 0; r < 8; ++r) {
        const int crow = row0 + r + half * 8;
        if (crow < nrows && ccol < D_DIM)
            Y[(long)crow * D_DIM + ccol] = acc[r];
    }
}

// ---------------------------------------------------------------------------
// One wave per edge: e = leaky_relu(||ent[h] + rel[et] - ent[t]||_2)
// float4 gathers (rows are 400B, 16B-aligned); lanes 0-24 active.
// Segment-max via uint atomicMax (valid: all e >= 0, bits are monotone).
// ---------------------------------------------------------------------------
__global__ void edge_score(const float* __restrict__ ent,
                           const float* __restrict__ rel,
                           const int* __restrict__ h,
                           const int* __restrict__ t,
                           const int* __restrict__ et,
                           float* __restrict__ e,
                           unsigned* __restrict__ seg_max, int n_edge) {
    const int wave = (blockIdx.x * blockDim.x + threadIdx.x) >> 5;
    const int lane = threadIdx.x & 31;
    if (wave >= n_edge) return;

    const int hh = h[wave], tt = t[wave], rr = et[wave];
    const float4* __restrict__ ph = (const float4*)(ent + (long)hh * D_DIM);
    const float4* __restrict__ pt = (const float4*)(ent + (long)tt * D_DIM);
    const float4* __restrict__ pr = (const float4*)(rel + (long)rr * D_DIM);

    float s = 0.0f;
    if (lane < D_VEC) {
        const float4 a = ph[lane];        // one global_load_b128 each
        const float4 b = pr[lane];
        const float4 c = pt[lane];
        const float dx = a.x + b.x - c.x;
        const float dy = a.y + b.y - c.y;
        const float dz = a.z + b.z - c.z;
        const float dw = a.w + b.w - c.w;
        s = fmaf(dx, dx, fmaf(dy, dy, fmaf(dz, dz, dw * dw)));
    }
#pragma unroll
    for (int off = 16; off > 0; off >>= 1)
        s += __shfl_xor(s, off, 32);

    if (lane == 0) {
        const float sc = sqrtf(s);
        const float lr = sc > 0.0f ? sc : ALPHA * sc;
        e[wave] = lr;
        atomicMax(seg_max + hh, __float_as_uint(lr));
    }
}

// ---------------------------------------------------------------------------
// One thread per edge: ex = exp(e - seg_max[h]); denom[h] += ex.
// ---------------------------------------------------------------------------
__global__ void edge_exp(const float* __restrict__ e,
                         const int* __restrict__ h,
                         const unsigned* __restrict__ seg_max,
                         float* __restrict__ ex,
                         float* __restrict__ denom, int n_edge) {
    const int i = blockIdx.x * blockDim.x + threadIdx.x;
    if (i >= n_edge) return;
    const int hh = h[i];
    const float m = __uint_as_float(seg_max[hh]);
    const float v = __expf(e[i] - m);
    ex[i] = v;
    atomicAdd(denom + hh, v);
}

// ---------------------------------------------------------------------------
// One wave per edge: out[h] += (ent[t] - rel[et]) * att
// float4 gathers, 4 scalar f32 atomics per active lane (hit L2).
// ---------------------------------------------------------------------------
__global__ void edge_msg(const float* __restrict__ ent,
                         const float* __restrict__ rel,
                         const int* __restrict__ h,
                         const int* __restrict__ t,
                         const int* __restrict__ et,
                         const float* __restrict__ ex,
                         const float* __restrict__ denom,
                         float* __restrict__ out, int n_edge) {
    const int wave = (blockIdx.x * blockDim.x + threadIdx.x) >> 5;
    const int lane = threadIdx.x & 31;
    if (wave >= n_edge) return;

    const int hh = h[wave], tt = t[wave], rr = et[wave];
    const float att = ex[wave] / (denom[hh] + 1e-16f);
    const float4* __restrict__ pt = (const float4*)(ent + (long)tt * D_DIM);
    const float4* __restrict__ pr = (const float4*)(rel + (long)rr * D_DIM);
    float* __restrict__ po = out + (long)hh * D_DIM;

    if (lane < D_VEC) {
        const float4 a = pt[lane];
        const float4 b = pr[lane];
        const int d = lane * 4;
        atomicAdd(po + d + 0, (a.x - b.x) * att);
        atomicAdd(po + d + 1, (a.y - b.y) * att);
        atomicAdd(po + d + 2, (a.z - b.z) * att);
        atomicAdd(po + d + 3, (a.w - b.w) * att);
    }
}

// ---------------------------------------------------------------------------
// ELU finish.
// ---------------------------------------------------------------------------
__global__ void elu_kernel(float* __restrict__ out, int n) {
    const int i = blockIdx.x * blockDim.x + threadIdx.x;
    if (i < n) {
        const float v = out[i];
        out[i] = v > 0.0f ? v : (__expf(v) - 1.0f);
    }
}

// ---------------------------------------------------------------------------
// Host launcher
// ---------------------------------------------------------------------------
extern "C" void kernel_launch(void* const* d_in, const int* in_sizes, int n_in,
                              void* d_out, int out_size, void* d_ws, size_t ws_size,
                              hipStream_t stream) {
    const float* E  = (const float*)d_in[0];   // [N_ENT, 100]
    const float* R  = (const float*)d_in[1];   // [N_REL, 100]
    const float* W  = (const float*)d_in[2];   // [100, 100]
    const int*   ei = (const int*)d_in[3];     // [2, N_EDGE]
    const int*   et = (const int*)d_in[4];     // [N_EDGE]

    const int n_ent  = in_sizes[0] / D_DIM;
    const int n_rel  = in_sizes[1] / D_DIM;
    const int n_edge = in_sizes[4];
    const int* h = ei;
    const int* t = ei + n_edge;

    // Workspace carving (256B aligned -> float4 row loads stay 16B aligned)
    auto align256 = [](size_t x) { return (x + 255) & ~(size_t)255; };
    char* ws = (char*)d_ws;
    size_t off = 0;
    float*    ent     = (float*)(ws + off); off += align256((size_t)n_ent * D_DIM * sizeof(float));
    float*    rel     = (float*)(ws + off); off += align256((size_t)n_rel * D_DIM * sizeof(float));
    float*    e       = (float*)(ws + off); off += align256((size_t)n_edge * sizeof(float));
    float*    ex      = (float*)(ws + off); off += align256((size_t)n_edge * sizeof(float));
    unsigned* seg_max = (unsigned*)(ws + off); off += align256((size_t)n_ent * sizeof(unsigned));
    float*    denom   = (float*)(ws + off); off += align256((size_t)n_ent * sizeof(float));
    (void)ws_size;

    float* out = (float*)d_out;
    const int n_out = n_ent * D_DIM;

    // 1) zero output + stats
    frgat_init<<<(n_out + 255) / 256, 256, 0, stream>>>(out, n_out, denom, seg_max, n_ent);

    // 2) projections via WMMA (one wave per 16x16 tile, 8 waves/block)
    {
        const int colTiles = (D_DIM + 15) / 16;
        const int tilesE = ((n_ent + 15) / 16) * colTiles;
        proj_wmma<<<(tilesE + 7) / 8, 256, 0, stream>>>(E, W, ent, n_ent);
        const int tilesR = ((n_rel + 15) / 16) * colTiles;
        proj_wmma<<<(tilesR + 7) / 8, 256, 0, stream>>>(R, W, rel, n_rel);
    }

    // 3) per-edge scores + segment max (one wave per edge)
    edge_score<<<(n_edge + 7) / 8, 256, 0, stream>>>(ent, rel, h, t, et, e, seg_max, n_edge);

    // 4) exp + denominator (one thread per edge)
    edge_exp<<<(n_edge + 255) / 256, 256, 0, stream>>>(e, h, seg_max, ex, denom, n_edge);

    // 5) weighted message scatter-add (one wave per edge)
    edge_msg<<<(n_edge + 7) / 8, 256, 0, stream>>>(ent, rel, h, t, et, ex, denom, out, n_edge);

    // 6) ELU
    elu_kernel<<<(n_out + 255) / 256, 256, 0, stream>>>(out, n_out);
}